// PETVarNet_4054449127428
// MI455X (gfx1250) — compile-verified
//
#include <hip/hip_runtime.h>

#define NB      4
#define NSUBS   8
#define NLORS   32768
#define RAYL    256
#define NN0     256
#define NN1     256
#define NVOXL   (NN0*NN1)
#define CHN     16

typedef float v2f __attribute__((ext_vector_type(2)));
typedef float v8f __attribute__((ext_vector_type(8)));

#if defined(__has_builtin)
# if __has_builtin(__builtin_amdgcn_global_load_async_to_lds_b32)
#  define HAVE_ASYNC_LDS 1
# endif
#endif
#ifndef HAVE_ASYNC_LDS
# define HAVE_ASYNC_LDS 0
#endif

__device__ __forceinline__ void wait_asynccnt0() {
#if defined(__has_builtin)
# if __has_builtin(__builtin_amdgcn_s_wait_asynccnt)
    __builtin_amdgcn_s_wait_asynccnt(0);
# else
    asm volatile("s_wait_asynccnt 0x0" ::: "memory");
# endif
#else
    asm volatile("s_wait_asynccnt 0x0" ::: "memory");
#endif
}

__device__ __forceinline__ void atomic_fadd(float* p, float v) {
    __hip_atomic_fetch_add(p, v, __ATOMIC_RELAXED, __HIP_MEMORY_SCOPE_AGENT);
}

// ---------------------------------------------------------------- zero
__global__ void k_zero(float* __restrict__ p, int n) {
    int i = blockIdx.x * blockDim.x + threadIdx.x;
    if (i < n) p[i] = 0.0f;
}

// ------------------------------------------------ fused fwd-proj + back-proj
// One wave32 per (batch, LOR): gather-dot -> butterfly reduce -> ratio ->
// immediate scatter (ray stream read exactly once per iteration).
__global__ __launch_bounds__(256) void k_fwdback(
    const float* __restrict__ x_in,
    const float* __restrict__ data,
    const float* __restrict__ mult,
    const float* __restrict__ contam,
    const float* __restrict__ norm,
    const int*   __restrict__ ray_idx,
    const float* __restrict__ pw,
    float*       __restrict__ back,
    int subset)
{
    int wave = (blockIdx.x * blockDim.x + threadIdx.x) >> 5;
    int lane = threadIdx.x & 31;
    int b = wave >> 15;              // NLORS = 2^15
    int l = wave & (NLORS - 1);
    size_t base = ((size_t)subset * NLORS + l) * RAYL;
    const float* img = x_in + (size_t)b * NVOXL;

    int   idx[8];
    float w[8];
    float acc = 0.0f;
#pragma unroll
    for (int k = 0; k < 8; ++k) {
        int o = lane + 32 * k;
        idx[k] = ray_idx[base + o];
        w[k]   = pw[base + o];
        // gfx1250 streaming prefetch hint for the ray stream (32 KB ahead)
        __builtin_prefetch(&ray_idx[base + o + 8192], 0, 1);
        __builtin_prefetch(&pw[base + o + 8192], 0, 1);
        acc += img[idx[k]] * w[k];
    }
#pragma unroll
    for (int off = 16; off > 0; off >>= 1)
        acc += __shfl_xor(acc, off, 32);

    size_t li = ((size_t)b * NSUBS + subset) * NLORS + l;
    float fwd      = norm[b] * acc;
    float m        = mult[li];
    float expected = m * fwd + contam[li];
    float ratio    = m * (1.0f - data[li] / expected);

    float* bb = back + (size_t)b * NVOXL;
#pragma unroll
    for (int k = 0; k < 8; ++k)
        atomic_fadd(&bb[idx[k]], w[k] * ratio);
}

// ---------------------------------------------------------------- finish
// x_data = back * x / adjoint_ones   (norm factors cancel)
__global__ void k_finish(const float* __restrict__ back,
                         const float* __restrict__ x_in,
                         const float* __restrict__ adj,
                         float*       __restrict__ xdata,
                         int subset)
{
    int i = blockIdx.x * blockDim.x + threadIdx.x;   // over NB*NVOXL
    int b = i >> 16;                                  // NVOXL = 2^16
    int v = i & (NVOXL - 1);
    size_t ai = ((size_t)b * NSUBS + subset) * NVOXL + v;
    xdata[i] = back[i] * x_in[i] / adj[ai];
}

// clamped LDS tile offset for a conv1 tap; taps >= 9 (K padding) and pixels of
// the discarded half-segment clamp in-bounds; their products hit zero B rows
// or discarded D columns, so values don't matter. Branch-free (v_min).
__device__ __forceinline__ int tap_off(int r, int base_c, int t) {
    int dy = t / 3 - 1;
    int dx = t % 3 - 1;
    int rl = r + 1 + dy;      rl = rl < 20 ? rl : 19;
    int cl = base_c + 1 + dx; cl = cl < 20 ? cl : 19;
    return rl * 21 + cl;
}

// ------------------------------------- fused conv1+relu -> LDS -> conv2 -> update
// 16x16 output tile / workgroup (8 waves). Full-precision f32 WMMA 16x16x4.
// Input tile staged to LDS (async-to-LDS), zero halo baked in. lds_h is 18x32
// so every conv1 output column has a legal slot -> unconditional ds_stores.
// Epilogue: x_out = relu(osem - x_data + nw * net(x))
__global__ __launch_bounds__(256) void k_conv_update(
    const float* __restrict__ x_in,
    const float* __restrict__ osem,
    const float* __restrict__ xdata,
    const float* __restrict__ w1, const float* __restrict__ b1,
    const float* __restrict__ w2, const float* __restrict__ b2,
    const float* __restrict__ net_weight,
    float*       __restrict__ x_out)
{
    __shared__ float lds_x[20 * 21];          // 20x20 input tile, stride 21
    __shared__ float lds_h[18 * 32 * CHN];    // 18 rows x 32 cols x 16 ch (36 KB)

    const int wave = threadIdx.x >> 5;
    const int lane = threadIdx.x & 31;
    const int half = lane >> 4;
    const int nl   = lane & 15;
    const int x0 = blockIdx.x * 16, y0 = blockIdx.y * 16, b = blockIdx.z;
    const float* img = x_in + (size_t)b * NVOXL;

    // ---- stage input tile (rows y0-2..y0+17, cols x0-2..x0+17) ----
    for (int e = threadIdx.x; e < 20 * 20; e += 256) {
        int ly = e / 20, lx = e % 20;
        int gy = y0 - 2 + ly, gx = x0 - 2 + lx;
        bool inb = ((unsigned)gy < NN0) && ((unsigned)gx < NN1);
#if HAVE_ASYNC_LDS
        if (inb) {
            __builtin_amdgcn_global_load_async_to_lds_b32(
                (int*)(img + gy * NN1 + gx),
                (int*)&lds_x[ly * 21 + lx],
                0, 0);
        } else {
            lds_x[ly * 21 + lx] = 0.0f;      // disjoint addresses: no hazard
        }
#else
        int cy = inb ? gy : 0, cx = inb ? gx : 0;
        float v = img[cy * NN1 + cx];        // clamped: always a legal load
        lds_x[ly * 21 + lx] = inb ? v : 0.0f;
#endif
    }
#if HAVE_ASYNC_LDS
    wait_asynccnt0();
#endif
    __syncthreads();

    // conv1 weight fragments: B1[k, c] = (k < 9) ? w1[c*9 + k] : 0
    v2f bw1[3];
#pragma unroll
    for (int ks = 0; ks < 3; ++ks) {
        int k0 = ks * 4 + 2 * half;
        bw1[ks].x = (k0     < 9) ? w1[nl * 9 + k0]     : 0.0f;
        bw1[ks].y = (k0 + 1 < 9) ? w1[nl * 9 + k0 + 1] : 0.0f;
    }
    float bias1 = b1[nl];

    // ---- conv1: h rows 0..17 via WMMA, 36 row-segments over 8 waves ----
    for (int seg = wave; seg < 36; seg += 8) {
        int r  = seg >> 1;                 // local h row 0..17
        int hs = seg & 1;                  // 16-wide half segment
        int bc = 16 * hs + nl;             // this lane's pixel (M = nl)

        v8f d;
#pragma unroll
        for (int j = 0; j < 8; ++j) d[j] = bias1;

#pragma unroll
        for (int ks = 0; ks < 3; ++ks) {
            int t0 = ks * 4 + 2 * half;
            v2f a;
            a.x = lds_x[tap_off(r, bc, t0)];
            a.y = lds_x[tap_off(r, bc, t0 + 1)];
            d = __builtin_amdgcn_wmma_f32_16x16x4_f32(
                    false, a, false, bw1[ks], (short)0, d, false, false);
        }
        // relu + store: VGPR j holds (M = j + 8*half, N = nl); all 32 cols
        // have slots in lds_h, so stores are unconditional (no exec churn).
#pragma unroll
        for (int j = 0; j < 8; ++j) {
            int lc = 16 * hs + j + 8 * half;
            float v = d[j];
            lds_h[(r * 32 + lc) * CHN + nl] = v > 0.0f ? v : 0.0f;
        }
    }
    __syncthreads();

    const float nw    = net_weight[0];
    const float bias2 = b2[0];

    // ---- conv2: K = 144 (tap-major, c-minor), WMMA column 0 = output ----
    for (int row = wave; row < 16; row += 8) {
        v8f d;
#pragma unroll
        for (int j = 0; j < 8; ++j) d[j] = 0.0f;

#pragma unroll
        for (int tap = 0; tap < 9; ++tap) {
            int dy = tap / 3 - 1, dx = tap % 3 - 1;
            int lr = row + 1 + dy;
            int lc = nl + 1 + dx;          // this lane's pixel (M = nl)
            const float* hp = &lds_h[(lr * 32 + lc) * CHN];
#pragma unroll
            for (int cs = 0; cs < 4; ++cs) {
                int c0 = cs * 4 + 2 * half;
                v2f a;                      // ds_load_b64 (c0 even)
                a.x = hp[c0];
                a.y = hp[c0 + 1];
                v2f bf; bf.x = 0.0f; bf.y = 0.0f;
                if (nl == 0) {              // only N==0 column carries weights
                    bf.x = w2[(c0)     * 9 + tap];
                    bf.y = w2[(c0 + 1) * 9 + tap];
                }
                d = __builtin_amdgcn_wmma_f32_16x16x4_f32(
                        false, a, false, bf, (short)0, d, false, false);
            }
        }
        // D column 0 lives in lanes 0 (M=j) and 16 (M=j+8)
        if (nl == 0) {
            int gy = y0 + row;
#pragma unroll
            for (int j = 0; j < 8; ++j) {
                int gx = x0 + j + 8 * half;
                size_t v = (size_t)b * NVOXL + (size_t)gy * NN1 + gx;
                float o   = bias2 + d[j];
                float val = osem[v] - xdata[v] + nw * o;
                x_out[v]  = val > 0.0f ? val : 0.0f;
            }
        }
    }
}

// ---------------------------------------------------------------- launch
extern "C" void kernel_launch(void* const* d_in, const int* in_sizes, int n_in,
                              void* d_out, int out_size, void* d_ws, size_t ws_size,
                              hipStream_t stream) {
    const float* osem   = (const float*)d_in[0];
    const float* data   = (const float*)d_in[1];
    const float* mult   = (const float*)d_in[2];
    const float* contam = (const float*)d_in[3];
    const float* adj    = (const float*)d_in[4];
    const float* norm   = (const float*)d_in[5];
    const int*   ridx   = (const int*)  d_in[6];
    const float* pw     = (const float*)d_in[7];
    const float* w1     = (const float*)d_in[8];
    const float* b1     = (const float*)d_in[9];
    const float* w2     = (const float*)d_in[10];
    const float* b2     = (const float*)d_in[11];
    const float* nw     = (const float*)d_in[12];
    float* out = (float*)d_out;

    float* ws    = (float*)d_ws;
    float* xA    = ws;
    float* xB    = ws + (size_t)NB * NVOXL;
    float* xdata = ws + (size_t)2 * NB * NVOXL;
    float* back  = ws + (size_t)3 * NB * NVOXL;

    const int subsets[4] = {0, 2, 4, 6};
    const float* xin = osem;                 // iteration 0 reads osem directly
    float* xout_seq[4] = {xA, xB, xA, out};  // ping-pong; last iter -> d_out

    for (int it = 0; it < 4; ++it) {
        int s = subsets[it];
        k_zero<<<(NB * NVOXL) / 256, 256, 0, stream>>>(back, NB * NVOXL);
        k_fwdback<<<(NB * NLORS) / 8, 256, 0, stream>>>(
            xin, data, mult, contam, norm, ridx, pw, back, s);
        k_finish<<<(NB * NVOXL) / 256, 256, 0, stream>>>(back, xin, adj, xdata, s);
        dim3 grid(NN1 / 16, NN0 / 16, NB);
        k_conv_update<<<grid, 256, 0, stream>>>(
            xin, osem, xdata, w1, b1, w2, b2, nw, xout_seq[it]);
        xin = xout_seq[it];
    }
}